// SpeakerClustering_56994216017947
// MI455X (gfx1250) — compile-verified
//
#include <hip/hip_runtime.h>

// Speaker clustering: kNN mask -> symmetrize -> Laplacian -> full sym eigh.
// N=4096 fp32. Eigensolver = blocked Householder tridiagonalization where the
// rank-2k trailing update (the O(N^3) GEMM part) runs on V_WMMA_F32_16X16X4_F32
// with 32x32-per-wave register blocking and symmetric (lower-block) evaluation,
// then Sturm bisection (all eigenvalues) + inverse iteration + back-transform
// (first 8 eigenvectors). Matrix (64MB) is L2-resident on MI455X (192MB L2).

#define N_DIM 4096
#define NB 16

typedef float v2f __attribute__((ext_vector_type(2)));
typedef float v8f __attribute__((ext_vector_type(8)));

__device__ __forceinline__ float blk_sum(float v, float* red) {
  int tid = threadIdx.x, nt = blockDim.x;
  red[tid] = v; __syncthreads();
  for (int s = nt >> 1; s > 0; s >>= 1) {
    if (tid < s) red[tid] += red[tid + s];
    __syncthreads();
  }
  float r = red[0]; __syncthreads();
  return r;
}

// ---------------- Step 1: per-row top-p threshold (bisection in LDS) --------
__global__ void k_topk_thresh(const float* __restrict__ A, const int* __restrict__ pptr,
                              float* __restrict__ th, int N) {
  __shared__ float srow[N_DIM];
  __shared__ float red[256];
  __shared__ float bc[2];
  int r = blockIdx.x, tid = threadIdx.x;
  const float* row = A + (size_t)r * N;
  float lmin = 3.4e38f, lmax = -3.4e38f;
  for (int c = tid; c < N; c += blockDim.x) {
    float x = row[c]; srow[c] = x;
    lmin = fminf(lmin, x); lmax = fmaxf(lmax, x);
  }
  red[tid] = lmin; __syncthreads();
  for (int s = 128; s > 0; s >>= 1) { if (tid < s) red[tid] = fminf(red[tid], red[tid + s]); __syncthreads(); }
  if (tid == 0) bc[0] = red[0];
  __syncthreads();
  red[tid] = lmax; __syncthreads();
  for (int s = 128; s > 0; s >>= 1) { if (tid < s) red[tid] = fmaxf(red[tid], red[tid + s]); __syncthreads(); }
  if (tid == 0) bc[1] = red[0];
  __syncthreads();
  float lo = bc[0], hi = bc[1];
  int p = *pptr;
  for (int it = 0; it < 40; ++it) {
    float mid = 0.5f * (lo + hi);
    float cnt = 0.f;
    for (int c = tid; c < N; c += blockDim.x) cnt += (srow[c] >= mid) ? 1.f : 0.f;
    float tot = blk_sum(cnt, red);
    if (tot >= (float)p) lo = mid; else hi = mid;
  }
  if (tid == 0) th[r] = lo;
}

// ---------------- Step 2: masked symmetrized affinity W ---------------------
// B[c,r] = A[c,r] * [A[r,c] >= th[r]];  W = 0.5*(B + B^T), zero diagonal.
__global__ void k_build_W(const float* __restrict__ A, const float* __restrict__ th,
                          float* __restrict__ Wm, int N) {
  size_t idx = (size_t)blockIdx.x * blockDim.x + threadIdx.x;
  size_t NN = (size_t)N * N;
  if (idx >= NN) return;
  int i = (int)(idx / N), j = (int)(idx % N);
  float aij = A[idx];
  float aji = A[(size_t)j * N + i];
  float w = 0.5f * (aij * ((aji >= th[j]) ? 1.f : 0.f) +
                    aji * ((aij >= th[i]) ? 1.f : 0.f));
  if (i == j) w = 0.f;
  Wm[idx] = w;
}

__global__ void k_rowsum(const float* __restrict__ Wm, float* __restrict__ D, int N) {
  __shared__ float red[256];
  int r = blockIdx.x;
  const float* row = Wm + (size_t)r * N;
  float acc = 0.f;
  for (int c = threadIdx.x; c < N; c += blockDim.x) acc += fabsf(row[c]);
  float tot = blk_sum(acc, red);
  if (threadIdx.x == 0) D[r] = tot;
}

__global__ void k_make_L(float* __restrict__ Wm, const float* __restrict__ D, int N) {
  size_t idx = (size_t)blockIdx.x * blockDim.x + threadIdx.x;
  if (idx >= (size_t)N * N) return;
  int i = (int)(idx / N), j = (int)(idx % N);
  Wm[idx] = (i == j) ? D[i] : -Wm[idx];
}

// ---------------- Step 3: blocked Householder tridiagonalization ------------
__global__ void k_zero_panels(float* __restrict__ U, float* __restrict__ W, int count) {
  int idx = blockIdx.x * blockDim.x + threadIdx.x;
  if (idx < count) { U[idx] = 0.f; W[idx] = 0.f; }
}

// A(j:N, j) -= U(:,0:i) * W(j,0:i)^T + W(:,0:i) * U(j,0:i)^T
__global__ void k_col_update(float* __restrict__ L, const float* __restrict__ U,
                             const float* __restrict__ W, int N, int j, int i) {
  int r = j + blockIdx.x * blockDim.x + threadIdx.x;
  if (r >= N) return;
  float acc = L[(size_t)r * N + j];
  const float* ur = U + (size_t)r * NB;
  const float* wr = W + (size_t)r * NB;
  const float* uj = U + (size_t)j * NB;
  const float* wj = W + (size_t)j * NB;
  for (int t = 0; t < i; ++t) acc -= ur[t] * wj[t] + wr[t] * uj[t];
  L[(size_t)r * N + j] = acc;
}

// Householder of column j (below diagonal). Records d[j], e[j], tau[j];
// stores v (v0=1) into the column, panel U[:,i] and contiguous vtmp.
__global__ void k_householder(float* __restrict__ L, float* __restrict__ U,
                              float* __restrict__ dv, float* __restrict__ ev,
                              float* __restrict__ tauv, float* __restrict__ vt,
                              int N, int j, int i) {
  __shared__ float red[1024];
  __shared__ float bc[1];
  int tid = threadIdx.x, nt = blockDim.x;
  if (tid == 0) dv[j] = L[(size_t)j * N + j];
  int mlen = N - 1 - j;
  if (mlen <= 0) { if (tid == 0) tauv[j] = 0.f; return; }
  float ss = 0.f;
  for (int k = 1 + tid; k < mlen; k += nt) {
    float x = L[(size_t)(j + 1 + k) * N + j];
    ss += x * x;
  }
  ss = blk_sum(ss, red);
  if (tid == 0) {
    float alpha = L[(size_t)(j + 1) * N + j];
    float beta, tau, scale;
    if (ss == 0.f) { beta = alpha; tau = 0.f; scale = 0.f; }
    else {
      float nrm = sqrtf(alpha * alpha + ss);
      beta = (alpha >= 0.f) ? -nrm : nrm;
      tau = (beta - alpha) / beta;
      scale = 1.0f / (alpha - beta);
    }
    ev[j] = beta; tauv[j] = tau; bc[0] = scale;
  }
  __syncthreads();
  float scale = bc[0];
  for (int k = tid; k < mlen; k += nt) {
    float x = L[(size_t)(j + 1 + k) * N + j];
    float v = (k == 0) ? 1.0f : x * scale;
    vt[k] = v;
    L[(size_t)(j + 1 + k) * N + j] = v;
    U[(size_t)(j + 1 + k) * NB + i] = v;
  }
}

// p = A(j+1:N, j+1:N) * v  (one block per row; matrix is L2-resident)
__global__ void k_symv(const float* __restrict__ L, const float* __restrict__ v,
                       float* __restrict__ p, int N, int j) {
  __shared__ float red[256];
  int mlen = N - 1 - j;
  int b = blockIdx.x;
  const float* row = L + (size_t)(j + 1 + b) * N + (j + 1);
  float acc = 0.f;
  for (int c = threadIdx.x; c < mlen; c += blockDim.x) {
    __builtin_prefetch(row + c + 4 * blockDim.x, 0, 1);
    acc = fmaf(row[c], v[c], acc);
  }
  float tot = blk_sum(acc, red);
  if (threadIdx.x == 0) p[b] = tot;
}

// w = tau*(p - U*(W^T v) - W*(U^T v));  w -= 0.5*tau*(w^T v)*v; store W[:,i]
__global__ void k_wfinal(const float* __restrict__ U, float* __restrict__ W,
                         const float* __restrict__ vt, const float* __restrict__ pt,
                         const float* __restrict__ tauv, int N, int j, int i) {
  __shared__ float red[1024];
  __shared__ float s1[NB], s2[NB];
  int tid = threadIdx.x, nt = blockDim.x;
  int mlen = N - 1 - j;
  for (int t = 0; t < i; ++t) {
    float a = 0.f, b = 0.f;
    for (int k = tid; k < mlen; k += nt) {
      int r = j + 1 + k; float v = vt[k];
      a += W[(size_t)r * NB + t] * v;
      b += U[(size_t)r * NB + t] * v;
    }
    float ta = blk_sum(a, red);
    float tb = blk_sum(b, red);
    if (tid == 0) { s1[t] = ta; s2[t] = tb; }
  }
  __syncthreads();
  float tau = tauv[j];
  for (int k = tid; k < mlen; k += nt) {
    int r = j + 1 + k;
    float w = pt[k];
    for (int t = 0; t < i; ++t)
      w -= U[(size_t)r * NB + t] * s1[t] + W[(size_t)r * NB + t] * s2[t];
    W[(size_t)r * NB + i] = tau * w;
  }
  __syncthreads();
  float cc = 0.f;
  for (int k = tid; k < mlen; k += nt) cc += W[(size_t)(j + 1 + k) * NB + i] * vt[k];
  cc = blk_sum(cc, red);
  float corr = 0.5f * tau * cc;
  for (int k = tid; k < mlen; k += nt)
    W[(size_t)(j + 1 + k) * NB + i] -= corr * vt[k];
}

// ---- fp32 WMMA rank-2k: A[k2:,k2:] -= U*W^T + W*U^T (K=16), symmetric ------
// One wave32 per 32x32 lower block: 2x2 register blocking (4 v8f accumulators),
// 32 chained v_wmma_f32_16x16x4_f32, fragments reused across the 2x2 tiles.
// Upper blocks written via the transposed (per-lane contiguous) mirror store.
__device__ __forceinline__ v8f wmma_f32(v2f a, v2f b, v8f c) {
  return __builtin_amdgcn_wmma_f32_16x16x4_f32(false, a, false, b, (short)0, c, false, false);
}

__device__ __forceinline__ void c_sub_store(float* __restrict__ L, int r0, int c0,
                                            int half, int lm, v8f acc) {
  float* p = L + (size_t)(r0 + 8 * half) * N_DIM + (c0 + lm);
#pragma unroll
  for (int g = 0; g < 8; ++g) { *p -= acc[g]; p += N_DIM; }
}

__device__ __forceinline__ void c_sub_store_mirror(float* __restrict__ L, int r0, int c0,
                                                   int half, int lm, v8f acc) {
  // element (r0+g+8*half, c0+lm) mirrored to (c0+lm, r0+g+8*half): contiguous in g
  float* p = L + (size_t)(c0 + lm) * N_DIM + (r0 + 8 * half);
#pragma unroll
  for (int g = 0; g < 8; ++g) p[g] -= acc[g];
}

__global__ void k_syr2k_wmma(float* __restrict__ L, const float* __restrict__ U,
                             const float* __restrict__ W, int k2) {
  int rb = blockIdx.x, cb = blockIdx.y;
  if (cb > rb) return;                       // symmetric: lower blocks only (uniform)
  int lane = threadIdx.x;
  int half = lane >> 4, lm = lane & 15;
  int r0 = k2 + rb * 32;
  int c0 = k2 + cb * 32;
  bool hr1 = (r0 + 16) < N_DIM;              // uniform edge predicates
  bool hc1 = (c0 + 16) < N_DIM;
  const float* ur0 = U + (size_t)(r0 + lm) * NB;
  const float* wr0 = W + (size_t)(r0 + lm) * NB;
  const float* ur1 = ur0 + 16 * NB;
  const float* wr1 = wr0 + 16 * NB;
  const float* uc0 = U + (size_t)(c0 + lm) * NB;
  const float* wc0 = W + (size_t)(c0 + lm) * NB;
  const float* uc1 = uc0 + 16 * NB;
  const float* wc1 = wc0 + 16 * NB;
  v8f a00 = {}, a01 = {}, a10 = {}, a11 = {};
#pragma unroll
  for (int s = 0; s < 4; ++s) {
    int kb = 4 * s + 2 * half;
    v2f aU0 = *(const v2f*)(ur0 + kb);
    v2f aW0 = *(const v2f*)(wr0 + kb);
    v2f bU0 = *(const v2f*)(uc0 + kb);
    v2f bW0 = *(const v2f*)(wc0 + kb);
    a00 = wmma_f32(aU0, bW0, a00);
    a00 = wmma_f32(aW0, bU0, a00);
    if (hr1) {
      v2f aU1 = *(const v2f*)(ur1 + kb);
      v2f aW1 = *(const v2f*)(wr1 + kb);
      a10 = wmma_f32(aU1, bW0, a10);
      a10 = wmma_f32(aW1, bU0, a10);
      if (hc1) {
        v2f bU1 = *(const v2f*)(uc1 + kb);
        v2f bW1 = *(const v2f*)(wc1 + kb);
        a01 = wmma_f32(aU0, bW1, a01);
        a01 = wmma_f32(aW0, bU1, a01);
        a11 = wmma_f32(aU1, bW1, a11);
        a11 = wmma_f32(aW1, bU1, a11);
      }
    } else if (hc1) {
      v2f bU1 = *(const v2f*)(uc1 + kb);
      v2f bW1 = *(const v2f*)(wc1 + kb);
      a01 = wmma_f32(aU0, bW1, a01);
      a01 = wmma_f32(aW0, bU1, a01);
    }
  }
  c_sub_store(L, r0, c0, half, lm, a00);
  if (hc1) c_sub_store(L, r0, c0 + 16, half, lm, a01);
  if (hr1) c_sub_store(L, r0 + 16, c0, half, lm, a10);
  if (hr1 && hc1) c_sub_store(L, r0 + 16, c0 + 16, half, lm, a11);
  if (rb != cb) {                            // mirror into the upper triangle
    c_sub_store_mirror(L, r0, c0, half, lm, a00);
    if (hc1) c_sub_store_mirror(L, r0, c0 + 16, half, lm, a01);
    if (hr1) c_sub_store_mirror(L, r0 + 16, c0, half, lm, a10);
    if (hr1 && hc1) c_sub_store_mirror(L, r0 + 16, c0 + 16, half, lm, a11);
  }
}

// ---------------- Step 4: eigenvalues via Sturm bisection -------------------
__global__ void k_gersh(const float* __restrict__ dv, const float* __restrict__ ev,
                        float* __restrict__ bnds, int N) {
  __shared__ float red[1024];
  int tid = threadIdx.x, nt = blockDim.x;
  float lo = 3.4e38f, hi = -3.4e38f;
  for (int i = tid; i < N; i += nt) {
    float r = 0.f;
    if (i > 0) r += fabsf(ev[i - 1]);
    if (i < N - 1) r += fabsf(ev[i]);
    lo = fminf(lo, dv[i] - r);
    hi = fmaxf(hi, dv[i] + r);
  }
  red[tid] = lo; __syncthreads();
  for (int s = nt >> 1; s > 0; s >>= 1) { if (tid < s) red[tid] = fminf(red[tid], red[tid + s]); __syncthreads(); }
  if (tid == 0) bnds[0] = red[0] - 1e-3f * fabsf(red[0]) - 1e-6f;
  __syncthreads();
  red[tid] = hi; __syncthreads();
  for (int s = nt >> 1; s > 0; s >>= 1) { if (tid < s) red[tid] = fmaxf(red[tid], red[tid + s]); __syncthreads(); }
  if (tid == 0) bnds[1] = red[0] + 1e-3f * fabsf(red[0]) + 1e-6f;
}

__global__ void k_sturm(const float* __restrict__ dv, const float* __restrict__ ev,
                        const float* __restrict__ bnds, float* __restrict__ lam, int N) {
  __shared__ float sd[N_DIM];
  __shared__ float se[N_DIM];
  for (int c = threadIdx.x; c < N; c += blockDim.x) {
    sd[c] = dv[c];
    se[c] = (c < N - 1) ? ev[c] : 0.f;
  }
  __syncthreads();
  int k = blockIdx.x * blockDim.x + threadIdx.x;
  float lo = bnds[0], hi = bnds[1];
  for (int it = 0; it < 60; ++it) {
    float mid = 0.5f * (lo + hi);
    int cnt = 0;
    float q = sd[0] - mid;
    cnt += (q < 0.f);
    for (int i = 1; i < N; ++i) {
      float dq = (q == 0.f) ? 1e-30f : q;
      q = (sd[i] - mid) - (se[i - 1] * se[i - 1]) / dq;
      cnt += (q < 0.f);
    }
    if (cnt <= k) lo = mid; else hi = mid;
  }
  lam[k] = 0.5f * (lo + hi);
}

// ---------------- Step 5: 8 eigenvectors: inverse iteration + back-transform
__global__ void k_inv_iter(const float* __restrict__ dv, const float* __restrict__ ev,
                           const float* __restrict__ lam, float* __restrict__ Y, int N) {
  __shared__ float cp[N_DIM];
  __shared__ float xx[N_DIM];
  __shared__ float red[256];
  int b = blockIdx.x, tid = threadIdx.x;
  float lambda = lam[b] + 1e-6f;
  for (int c = tid; c < N; c += blockDim.x)
    xx[c] = 1.0f + 0.001f * (float)((c * 37 + b * 101) & 63);
  __syncthreads();
  for (int iter = 0; iter < 2; ++iter) {
    if (tid == 0) {
      float bb = dv[0] - lambda;
      if (fabsf(bb) < 1e-20f) bb = 1e-20f;
      cp[0] = ev[0] / bb;
      xx[0] = xx[0] / bb;
      for (int i = 1; i < N; ++i) {
        float a = ev[i - 1];
        float m = (dv[i] - lambda) - a * cp[i - 1];
        if (fabsf(m) < 1e-20f) m = 1e-20f;
        float c2 = (i < N - 1) ? ev[i] : 0.f;
        cp[i] = c2 / m;
        xx[i] = (xx[i] - a * xx[i - 1]) / m;
      }
      for (int i = N - 2; i >= 0; --i) xx[i] -= cp[i] * xx[i + 1];
    }
    __syncthreads();
    float ss = 0.f;
    for (int c = tid; c < N; c += blockDim.x) ss += xx[c] * xx[c];
    ss = blk_sum(ss, red);
    float inv = rsqrtf(fmaxf(ss, 1e-30f));
    for (int c = tid; c < N; c += blockDim.x) xx[c] *= inv;
    __syncthreads();
  }
  for (int c = tid; c < N; c += blockDim.x) Y[(size_t)b * N + c] = xx[c];
}

__global__ void k_backtransform(const float* __restrict__ Y, const float* __restrict__ L,
                                const float* __restrict__ tauv, float* __restrict__ X, int N) {
  __shared__ float xx[N_DIM];
  __shared__ float red[1024];
  int b = blockIdx.x, tid = threadIdx.x, nt = blockDim.x;
  for (int c = tid; c < N; c += nt) xx[c] = Y[(size_t)b * N + c];
  __syncthreads();
  for (int j = N - 2; j >= 0; --j) {
    float tau = tauv[j];
    if (tau != 0.f) {
      int mlen = N - 1 - j;
      float dp = 0.f;
      for (int k = tid; k < mlen; k += nt) {
        float v = L[(size_t)(j + 1 + k) * N + j];
        dp += v * xx[j + 1 + k];
      }
      dp = blk_sum(dp, red);
      float c = tau * dp;
      for (int k = tid; k < mlen; k += nt) {
        float v = L[(size_t)(j + 1 + k) * N + j];
        xx[j + 1 + k] -= c * v;
      }
      __syncthreads();
    }
  }
  for (int c = tid; c < N; c += nt) X[(size_t)b * N + c] = xx[c];
}

// ---------------- Step 6: eigengap + output packing -------------------------
__global__ void k_finalize(const float* __restrict__ lam, const float* __restrict__ X,
                           const int* __restrict__ msptr, float* __restrict__ out, int N) {
  int idx = blockIdx.x * blockDim.x + threadIdx.x;
  int total = 1 + N + 8 * N;
  if (idx >= total) return;
  if (idx == 0) {
    int ms = *msptr; if (ms > 8) ms = 8;
    int best = 0; float bg = lam[1] - lam[0];
    for (int t = 1; t < ms; ++t) {
      float g = lam[t + 1] - lam[t];
      if (g > bg) { bg = g; best = t; }
    }
    out[0] = (float)(best + 1);
  } else if (idx <= N) {
    out[idx] = lam[idx - 1];
  } else {
    int q = idx - 1 - N;
    int r = q >> 3, c = q & 7;
    out[idx] = X[(size_t)c * N + r];
  }
}

extern "C" void kernel_launch(void* const* d_in, const int* in_sizes, int n_in,
                              void* d_out, int out_size, void* d_ws, size_t ws_size,
                              hipStream_t stream) {
  (void)in_sizes; (void)n_in; (void)out_size; (void)ws_size;
  const int N = N_DIM;
  const float* A = (const float*)d_in[0];
  const int* pptr = (const int*)d_in[1];
  const int* msptr = (const int*)d_in[2];
  float* ws = (float*)d_ws;
  size_t NN = (size_t)N * N;
  float* Lm   = ws;                     // N*N
  float* th   = ws + NN;                // N
  float* Dv   = th + N;                 // N
  float* Up   = Dv + N;                 // N*NB
  float* Wp   = Up + (size_t)N * NB;    // N*NB
  float* dv   = Wp + (size_t)N * NB;    // N
  float* ev   = dv + N;                 // N
  float* tv   = ev + N;                 // N
  float* vt   = tv + N;                 // N
  float* pt   = vt + N;                 // N
  float* lam  = pt + N;                 // N
  float* bnds = lam + N;                // 16
  float* Yt   = bnds + 16;              // 8*N
  float* Xf   = Yt + (size_t)8 * N;     // 8*N

  int nblk = (int)((NN + 255) / 256);
  k_topk_thresh<<<N, 256, 0, stream>>>(A, pptr, th, N);
  k_build_W<<<nblk, 256, 0, stream>>>(A, th, Lm, N);
  k_rowsum<<<N, 256, 0, stream>>>(Lm, Dv, N);
  k_make_L<<<nblk, 256, 0, stream>>>(Lm, Dv, N);

  for (int k0 = 0; k0 < N; k0 += NB) {
    k_zero_panels<<<(N * NB + 255) / 256, 256, 0, stream>>>(Up, Wp, N * NB);
    for (int i = 0; i < NB; ++i) {
      int j = k0 + i;
      if (i > 0)
        k_col_update<<<(N - j + 255) / 256, 256, 0, stream>>>(Lm, Up, Wp, N, j, i);
      k_householder<<<1, 1024, 0, stream>>>(Lm, Up, dv, ev, tv, vt, N, j, i);
      int mlen = N - 1 - j;
      if (mlen > 0) {
        k_symv<<<mlen, 256, 0, stream>>>(Lm, vt, pt, N, j);
        k_wfinal<<<1, 1024, 0, stream>>>(Up, Wp, vt, pt, tv, N, j, i);
      }
    }
    int k2 = k0 + NB;
    if (k2 < N) {
      int nt16 = (N - k2) / 16;           // trailing size in 16-tiles
      int nb32 = (nt16 + 1) / 2;          // 32x32 wave blocks
      dim3 grid(nb32, nb32);
      k_syr2k_wmma<<<grid, 32, 0, stream>>>(Lm, Up, Wp, k2);
    }
  }

  k_gersh<<<1, 1024, 0, stream>>>(dv, ev, bnds, N);
  k_sturm<<<N / 256, 256, 0, stream>>>(dv, ev, bnds, lam, N);
  k_inv_iter<<<8, 256, 0, stream>>>(dv, ev, lam, Yt, N);
  k_backtransform<<<8, 1024, 0, stream>>>(Yt, Lm, tv, Xf, N);
  int total = 1 + N + 8 * N;
  k_finalize<<<(total + 255) / 256, 256, 0, stream>>>(lam, Xf, msptr, (float*)d_out, N);
}